// TVDenoiser_37580963840013
// MI455X (gfx1250) — compile-verified
//
#include <hip/hip_runtime.h>
#include <cstdint>
#include <cstddef>

// ---- problem constants (match reference) ----
#define HH 512
#define WW 512
#define NB 8
#define NPIX ((size_t)NB * HH * WW)   // 2,097,152
#define NITER 40

constexpr float TAU     = 0.01f;
constexpr float RHO     = 1.99f;
constexpr float SIGMA   = 12.5f;            // 1/TAU/8
constexpr float INV_THS = 10.0f;            // 1/THS
constexpr float INV_1PT = 1.0f / 1.01f;     // 1/(1+TAU)
constexpr float CRITV   = 1e-5f;

// ---- CDNA5 async global->LDS path (guarded; fallback = sync LDS staging) ----
#if defined(__AMDGCN__) && __has_builtin(__builtin_amdgcn_global_load_async_to_lds_b128)
#define USE_ASYNC_LDS 1
typedef int v4i __attribute__((vector_size(4 * sizeof(int))));
typedef __attribute__((address_space(1))) v4i as1_v4i;   // global
typedef __attribute__((address_space(3))) v4i as3_v4i;   // LDS
#endif

__device__ __forceinline__ void copy16_to_lds(void* lds_dst, const void* gsrc) {
#ifdef USE_ASYNC_LDS
  __builtin_amdgcn_global_load_async_to_lds_b128(
      (as1_v4i*)gsrc,        // const-strip + addrspacecast generic->global
      (as3_v4i*)lds_dst,     // addrspacecast generic->LDS
      /*offset=*/0, /*cpol=*/0);
#else
  *(float4*)lds_dst = *(const float4*)gsrc;
#endif
}

__device__ __forceinline__ void staging_join() {
#ifdef USE_ASYNC_LDS
#if __has_builtin(__builtin_amdgcn_s_wait_asynccnt)
  __builtin_amdgcn_s_wait_asynccnt(0);
#else
  asm volatile("s_wait_asynccnt 0" ::: "memory");
#endif
#endif
  __syncthreads();
}

// ============================ init ============================
__global__ void __launch_bounds__(256)
tv_init(const float* __restrict__ y, float* __restrict__ x2,
        float* __restrict__ u0, float* __restrict__ u1, int* __restrict__ flag) {
  size_t i = ((size_t)blockIdx.x * 256 + threadIdx.x) * 4;
  if (i < NPIX) {
    float4 yv = *(const float4*)(y + i);
    *(float4*)(x2 + i) = yv;
    *(float4*)(u0 + i) = make_float4(0.f, 0.f, 0.f, 0.f);
    *(float4*)(u1 + i) = make_float4(0.f, 0.f, 0.f, 0.f);
  }
  if (i == 0) *flag = 0;
}

// ============================ phase A ============================
// x   = (x2 - TAU*div(u) + TAU*y)/(1+TAU)
// xbar= 2x - x2 ; x2 <- x2 + RHO*(x - x2) ; per-block partial sums for rel-norm
__global__ void __launch_bounds__(256)
tv_phaseA(const float* __restrict__ y, float* __restrict__ x2,
          const float* __restrict__ u0, const float* __restrict__ u1,
          float* __restrict__ xbar, float* __restrict__ partials,
          const int* __restrict__ flag) {
  if (*flag) return;

  __shared__ __align__(16) float su0[33][36];   // rows h0-1..h0+31, cols w0-4..w0+31
  __shared__ __align__(16) float su1[33][36];
  __shared__ float wred[8][2];

  const int h0 = blockIdx.y * 32;
  const int w0 = blockIdx.x * 32;
  const int tid = threadIdx.x;
  const size_t plane = (size_t)blockIdx.z * (HH * WW);

  // stage u0,u1 tile+halo: 2 planes * 33 rows * 9 x b128 chunks = 594 copies
  for (int c = tid; c < 2 * 33 * 9; c += 256) {
    int pl  = c / (33 * 9);
    int rem = c - pl * (33 * 9);
    int r   = rem / 9;
    int cc  = rem - r * 9;
    int gr  = h0 - 1 + r;        // -1..511
    int gc  = w0 - 4 + cc * 4;   // -4..508 (16B aligned)
    if (gr >= 0 && gc >= 0) {
      const float* src = (pl == 0 ? u0 : u1) + plane + (size_t)gr * WW + gc;
      float* dst = (pl == 0 ? &su0[r][cc * 4] : &su1[r][cc * 4]);
      copy16_to_lds(dst, src);
    }
  }
  staging_join();

  // each thread: 1 row, 4 consecutive columns (b128 global traffic)
  const int ly = tid >> 3;             // 0..31
  const int cg = (tid & 7) * 4;        // 0,4,...,28
  const int h  = h0 + ly;
  const int lr = ly + 1;               // LDS row for h
  const int lc = cg + 4;               // LDS col for first w
  const size_t idx = plane + (size_t)h * WW + (w0 + cg);

  float4 c0v = *(const float4*)&su0[lr][lc];
  float4 a0v = *(const float4*)&su0[lr - 1][lc];
  float4 c1v = *(const float4*)&su1[lr][lc];
  float  a1m = su1[lr][lc - 1];
  float4 xov = *(const float4*)(x2 + idx);
  float4 yvv = *(const float4*)(y + idx);

  float c0a[4] = {c0v.x, c0v.y, c0v.z, c0v.w};
  float a0a[4] = {a0v.x, a0v.y, a0v.z, a0v.w};
  float c1a[4] = {c1v.x, c1v.y, c1v.z, c1v.w};
  float a1a[4] = {a1m,   c1v.x, c1v.y, c1v.z};
  float xoa[4] = {xov.x, xov.y, xov.z, xov.w};
  float yva[4] = {yvv.x, yvv.y, yvv.z, yvv.w};
  float xba[4], xna[4];

  float dsq = 0.0f, nsq = 0.0f;
#pragma unroll
  for (int j = 0; j < 4; ++j) {
    const int w = w0 + cg + j;
    // divergence (adjoint of forward diff); halo garbage is gated out
    float divu = (h < HH - 1 ? -c0a[j] : 0.0f) + (h > 0 ? a0a[j] : 0.0f)
               + (w < WW - 1 ? -c1a[j] : 0.0f) + (w > 0 ? a1a[j] : 0.0f);
    float x  = (xoa[j] - TAU * divu + TAU * yva[j]) * INV_1PT;
    xba[j] = 2.0f * x - xoa[j];
    xna[j] = xoa[j] + RHO * (x - xoa[j]);
    float d = xoa[j] - xna[j];
    dsq += d * d;
    float nn = xna[j] + 1e-12f;
    nsq += nn * nn;
  }
  *(float4*)(xbar + idx) = make_float4(xba[0], xba[1], xba[2], xba[3]);
  *(float4*)(x2 + idx)   = make_float4(xna[0], xna[1], xna[2], xna[3]);

  // deterministic block reduction: wave32 shuffles + fixed-order wave sum
#pragma unroll
  for (int off = 16; off > 0; off >>= 1) {
    dsq += __shfl_down(dsq, off, 32);
    nsq += __shfl_down(nsq, off, 32);
  }
  if ((tid & 31) == 0) { wred[tid >> 5][0] = dsq; wred[tid >> 5][1] = nsq; }
  __syncthreads();
  if (tid == 0) {
    float D = 0.0f, Nn = 0.0f;
#pragma unroll
    for (int i = 0; i < 8; ++i) { D += wred[i][0]; Nn += wred[i][1]; }
    int bid = (blockIdx.z * gridDim.y + blockIdx.y) * gridDim.x + blockIdx.x;
    partials[2 * bid]     = D;
    partials[2 * bid + 1] = Nn;
  }
}

// ============================ phase B ============================
// v = u + SIGMA*grad(xbar); u <- u + RHO*(v/max(|v|/THS,1) - u)
__global__ void __launch_bounds__(256)
tv_phaseB(const float* __restrict__ xbar, float* __restrict__ u0,
          float* __restrict__ u1, const int* __restrict__ flag) {
  if (*flag) return;

  __shared__ __align__(16) float sxb[33][36];  // rows h0..h0+32, cols w0..w0+35

  const int h0 = blockIdx.y * 32;
  const int w0 = blockIdx.x * 32;
  const int tid = threadIdx.x;
  const size_t plane = (size_t)blockIdx.z * (HH * WW);

  for (int c = tid; c < 33 * 9; c += 256) {
    int r  = c / 9;
    int cc = c - r * 9;
    int gr = h0 + r;          // h0..h0+32
    int gc = w0 + cc * 4;     // w0..w0+32 (mult of 4)
    if (gr < HH && gc < WW) {
      copy16_to_lds(&sxb[r][cc * 4], xbar + plane + (size_t)gr * WW + gc);
    }
  }
  staging_join();

  const int ly = tid >> 3;             // 0..31
  const int cg = (tid & 7) * 4;        // 0..28
  const int h  = h0 + ly;
  const size_t idx = plane + (size_t)h * WW + (w0 + cg);

  float4 xcv = *(const float4*)&sxb[ly][cg];
  float4 dnv = *(const float4*)&sxb[ly + 1][cg];
  float  rt3 = sxb[ly][cg + 4];
  float4 p0v = *(const float4*)(u0 + idx);
  float4 p1v = *(const float4*)(u1 + idx);

  float xca[4] = {xcv.x, xcv.y, xcv.z, xcv.w};
  float dna[4] = {dnv.x, dnv.y, dnv.z, dnv.w};
  float rta[4] = {xcv.y, xcv.z, xcv.w, rt3};
  float p0a[4] = {p0v.x, p0v.y, p0v.z, p0v.w};
  float p1a[4] = {p1v.x, p1v.y, p1v.z, p1v.w};
  float q0a[4], q1a[4];

#pragma unroll
  for (int j = 0; j < 4; ++j) {
    const int w = w0 + cg + j;
    float dh = (h < HH - 1) ? (dna[j] - xca[j]) : 0.0f;
    float dw = (w < WW - 1) ? (rta[j] - xca[j]) : 0.0f;
    float v0 = p0a[j] + SIGMA * dh;
    float v1 = p1a[j] + SIGMA * dw;
    float mag = sqrtf(v0 * v0 + v1 * v1);
    float den = fmaxf(mag * INV_THS, 1.0f);
    float inv = 1.0f / den;
    q0a[j] = p0a[j] + RHO * (v0 * inv - p0a[j]);
    q1a[j] = p1a[j] + RHO * (v1 * inv - p1a[j]);
  }
  *(float4*)(u0 + idx) = make_float4(q0a[0], q0a[1], q0a[2], q0a[3]);
  *(float4*)(u1 + idx) = make_float4(q1a[0], q1a[1], q1a[2], q1a[3]);
}

// ============================ convergence check ============================
// Deterministic fixed-order sum of 2048 block partials; reproduces
// reference early-stop: after body with it=i+1, stop if it>=3 && rel<CRIT.
__global__ void __launch_bounds__(256)
tv_check(const float* __restrict__ partials, int* __restrict__ flag, int it) {
  __shared__ float sD[256];
  __shared__ float sN[256];
  const int tid = threadIdx.x;
  float D = 0.0f, Nn = 0.0f;
  for (int i = tid; i < 2048; i += 256) {
    D  += partials[2 * i];
    Nn += partials[2 * i + 1];
  }
  sD[tid] = D; sN[tid] = Nn;
  __syncthreads();
  for (int s = 128; s > 0; s >>= 1) {
    if (tid < s) { sD[tid] += sD[tid + s]; sN[tid] += sN[tid + s]; }
    __syncthreads();
  }
  if (tid == 0) {
    float rel = sqrtf(sD[0]) / sqrtf(sN[0]);
    if ((it + 1) >= 3 && rel < CRITV) *flag = 1;
  }
}

// ============================ launch ============================
extern "C" void kernel_launch(void* const* d_in, const int* in_sizes, int n_in,
                              void* d_out, int out_size, void* d_ws, size_t ws_size,
                              hipStream_t stream) {
  (void)in_sizes; (void)n_in; (void)out_size; (void)ws_size;
  const float* y = (const float*)d_in[0];
  float* x2 = (float*)d_out;                   // x2 lives in the output buffer
  float* ws = (float*)d_ws;
  float* u0       = ws;                        // 8 MB
  float* u1       = ws + NPIX;                 // 8 MB
  float* xbar     = ws + 2 * NPIX;             // 8 MB
  float* partials = ws + 3 * NPIX;             // 2048*2 floats
  int*   flag     = (int*)(ws + 3 * NPIX + 8192);

  tv_init<<<(unsigned)((NPIX / 4 + 255) / 256), 256, 0, stream>>>(y, x2, u0, u1, flag);

  dim3 grid(WW / 32, HH / 32, NB);             // 16 x 16 x 8 = 2048 blocks
  for (int it = 0; it < NITER; ++it) {
    tv_phaseA<<<grid, 256, 0, stream>>>(y, x2, u0, u1, xbar, partials, flag);
    tv_phaseB<<<grid, 256, 0, stream>>>(xbar, u0, u1, flag);
    if (it < NITER - 1) {
      tv_check<<<1, 256, 0, stream>>>(partials, flag, it);
    }
  }
}